// S6_68109591380838
// MI455X (gfx1250) — compile-verified
//
#include <hip/hip_runtime.h>
#include <math.h>
#include <stdint.h>

// Problem constants (match reference)
#define D_MODEL  1024
#define D_STATE  16
#define D_CONV   4
#define D_INNER  2048
#define DT_RANK  64
#define BATCH    2
#define SEQLEN   2048
#define NROWS    (BATCH * SEQLEN)   // 4096 token rows

typedef float v2f __attribute__((ext_vector_type(2)));
typedef float v8f __attribute__((ext_vector_type(8)));

__device__ __forceinline__ float silu_f(float x) {
    return x / (1.0f + __expf(-x));
}

// Async DMA: global -> LDS, 16 bytes per lane, tracked by ASYNCcnt.
__device__ __forceinline__ void async_g2l_b128(const float* gptr, const float* lptr) {
    const unsigned lds_addr = (unsigned)(uintptr_t)lptr;   // LDS aperture: offset = addr[31:0]
    asm volatile("global_load_async_to_lds_b128 %0, %1, off"
                 :: "v"(lds_addr), "v"(gptr) : "memory");
}
__device__ __forceinline__ void wait_async0() {
    asm volatile("s_wait_asynccnt 0x0" ::: "memory");
}

// ---------------------------------------------------------------------------
// Double-buffered f32 WMMA GEMM: C[M,N] = A[M,K] * B[K,N], row-major, strided.
// Block = 128 threads (4 waves stacked in M). Block tile = 64(M) x 16*NT(N).
// A tiles: GLOBAL_LOAD_ASYNC_TO_LDS_B128 (ASYNCcnt DMA), row-major [row][36]
//   -> A fragment = one 8B LDS read (K pair contiguous).
// B tiles: software-pipelined VGPR staging into *transposed* LDS [col][34]
//   -> B fragment = one 8B LDS read; compiler fuses pairs into ds_load_2addr_b64
//      feeding v_wmma operands directly (no register shuffles).
// Requirements (satisfied at launch sites): M % 64 == 0, K % 32 == 0,
// N % (16*NT) == 0, all row strides multiples of 4 floats.
// ---------------------------------------------------------------------------
template<int NT>
__global__ void wmma_gemm_lds(const float* __restrict__ A,
                              const float* __restrict__ B,
                              float* __restrict__ C,
                              int M, int N, int K,
                              int lda, int ldb, int ldc)
{
    constexpr int BN    = 16 * NT;   // block N width
    constexpr int KB    = 32;        // K block
    constexpr int PADA  = 36;        // sA k-stride (floats): 16B-aligned, bank-spread
    constexpr int PADBT = 34;        // sBt k-stride (floats): 8B-aligned, bank-spread

    __shared__ float sA [2][64 * PADA];   // [buf][row][k]
    __shared__ float sBt[2][BN * PADBT];  // [buf][col][k]  (transposed B)

    const int tid    = threadIdx.x;
    const int lane   = tid & 31;
    const int waveId = tid >> 5;        // 0..3
    const int lid    = lane & 15;
    const int hi     = lane >> 4;       // 0/1 lane half

    const int mBlock = blockIdx.y * 64;
    const int nBase  = blockIdx.x * BN;

    // B staging addresses for this thread (constant across K-blocks)
    int brow[NT], bc4[NT];
    #pragma unroll
    for (int j = 0; j < NT; ++j) {
        const int f = tid + j * 128;            // 0 .. 128*NT-1
        brow[j] = f / (BN / 4);                 // k 0..31
        bc4[j]  = (f % (BN / 4)) * 4;           // col offset (mult of 4)
    }

    v8f acc[NT];
    #pragma unroll
    for (int t = 0; t < NT; ++t) acc[t] = (v8f){0,0,0,0,0,0,0,0};

    auto stageA = [&](int k0, int buf) {
        #pragma unroll
        for (int j = 0; j < 4; ++j) {           // 64 rows x 32 k = 512 float4
            const int f   = tid + j * 128;
            const int row = f >> 3;
            const int kq  = (f & 7) << 2;
            async_g2l_b128(A + (size_t)(mBlock + row) * lda + (k0 + kq),
                           &sA[buf][row * PADA + kq]);
        }
    };
    auto loadB = [&](int k0, float4* reg) {
        #pragma unroll
        for (int j = 0; j < NT; ++j)
            reg[j] = *(const float4*)(B + (size_t)(k0 + brow[j]) * ldb + (nBase + bc4[j]));
    };
    auto storeB = [&](const float4* reg, int buf) {
        #pragma unroll
        for (int j = 0; j < NT; ++j) {
            sBt[buf][(bc4[j] + 0) * PADBT + brow[j]] = reg[j].x;
            sBt[buf][(bc4[j] + 1) * PADBT + brow[j]] = reg[j].y;
            sBt[buf][(bc4[j] + 2) * PADBT + brow[j]] = reg[j].z;
            sBt[buf][(bc4[j] + 3) * PADBT + brow[j]] = reg[j].w;
        }
    };

    // Prologue: stage K-block 0
    float4 bReg[NT];
    stageA(0, 0);
    loadB(0, bReg);
    storeB(bReg, 0);
    wait_async0();
    __syncthreads();

    int cur = 0;
    for (int k0 = 0; k0 < K; k0 += KB) {
        const bool hasNext = (k0 + KB) < K;
        if (hasNext) {
            stageA(k0 + KB, cur ^ 1);           // DMA overlaps compute
            loadB(k0 + KB, bReg);               // global loads in flight over compute
        }

        const float* aPtr = &sA [cur][(waveId * 16 + lid) * PADA + hi * 2];
        const float* bPtr = &sBt[cur][lid * PADBT + hi * 2];

        #pragma unroll
        for (int kk = 0; kk < KB; kk += 4) {
            const v2f aFrag = *(const v2f*)(aPtr + kk);
            #pragma unroll
            for (int t = 0; t < NT; ++t) {
                const v2f bFrag = *(const v2f*)(bPtr + t * 16 * PADBT + kk);
                acc[t] = __builtin_amdgcn_wmma_f32_16x16x4_f32(
                            false, aFrag, false, bFrag,
                            (short)0, acc[t], false, false);
            }
        }

        if (hasNext) storeB(bReg, cur ^ 1);     // one loadcnt wait, then ds stores
        wait_async0();                          // A DMA for next block landed
        __syncthreads();                        // all waves done reading 'cur'
        cur ^= 1;
    }

    // Store D: VGPR r -> row mBase + r + 8*hi, col nBase + 16*t + lid
    const int mBase = mBlock + waveId * 16;
    #pragma unroll
    for (int t = 0; t < NT; ++t) {
        const int col = nBase + t * 16 + lid;
        #pragma unroll
        for (int r = 0; r < 8; ++r) {
            const int row = mBase + r + hi * 8;
            C[(size_t)row * ldc + col] = acc[t][r];
        }
    }
}

// ---------------------------------------------------------------------------
// Depthwise causal conv (k=4) over seq dim + SiLU.
// ---------------------------------------------------------------------------
__global__ void conv_silu_kernel(const float* __restrict__ xr,
                                 const float* __restrict__ conv_w,
                                 const float* __restrict__ conv_b,
                                 float* __restrict__ u)
{
    const int d = blockIdx.x * blockDim.x + threadIdx.x;   // 0..2047
    const int l = blockIdx.y;
    const int b = blockIdx.z;

    float acc = conv_b[d];
    #pragma unroll
    for (int j = 0; j < D_CONV; ++j) {
        const int ls = l + j - (D_CONV - 1);
        if (ls >= 0) {
            acc += conv_w[d * D_CONV + j] *
                   xr[((size_t)b * SEQLEN + ls) * (2 * D_INNER) + d];
        }
    }
    u[((size_t)b * SEQLEN + l) * D_INNER + d] = silu_f(acc);
}

// ---------------------------------------------------------------------------
// delta = softplus(delta_pre + b_dt) in place, over (NROWS, D_INNER)
// ---------------------------------------------------------------------------
__global__ void softplus_bias_kernel(float* __restrict__ delta,
                                     const float* __restrict__ b_dt,
                                     int total)
{
    const int i = blockIdx.x * blockDim.x + threadIdx.x;
    if (i >= total) return;
    const int d = i & (D_INNER - 1);
    const float x = delta[i] + b_dt[d];
    delta[i] = (x > 20.0f) ? x : log1pf(__expf(x));
}

// ---------------------------------------------------------------------------
// Selective scan: one channel per thread, sequential over L.
// B/C (shared across all channels) staged via async DMA in 64-step chunks:
// sBC[lt][0:16] = B_t, sBC[lt][16:32] = C_t  (contiguous 32 floats per step).
// ---------------------------------------------------------------------------
#define SCAN_TS 64

__global__ void scan_kernel(const float* __restrict__ delta,
                            const float* __restrict__ dbc,
                            const float* __restrict__ u,
                            const float* __restrict__ A_log,
                            float* __restrict__ ys)
{
    const int d = blockIdx.x * blockDim.x + threadIdx.x;   // channel 0..2047
    const int b = blockIdx.y;

    __shared__ float sBC[SCAN_TS][2 * D_STATE];            // 8 KB

    float a[D_STATE];
    float h[D_STATE];
    #pragma unroll
    for (int n = 0; n < D_STATE; ++n) {
        a[n] = -__expf(A_log[(size_t)d * D_STATE + n]);
        h[n] = 0.0f;
    }

    for (int l0 = 0; l0 < SEQLEN; l0 += SCAN_TS) {
        // 64 steps * 32 floats = 512 float4; 256 threads -> 2 async b128 each
        #pragma unroll
        for (int j = 0; j < 2; ++j) {
            const int f  = threadIdx.x + j * 256;   // 0..511
            const int lt = f >> 3;                  // timestep in chunk
            const int c4 = (f & 7) << 2;            // 0,4,..,28
            async_g2l_b128(dbc + ((size_t)b * SEQLEN + l0 + lt) * 96 + DT_RANK + c4,
                           &sBC[lt][c4]);
        }
        wait_async0();
        __syncthreads();

        for (int lt = 0; lt < SCAN_TS; ++lt) {
            const size_t idx = ((size_t)b * SEQLEN + l0 + lt) * D_INNER + d;
            const float dt = delta[idx];
            const float uv = u[idx];
            const float dtu = dt * uv;
            float y = 0.0f;
            #pragma unroll
            for (int n = 0; n < D_STATE; ++n) {
                const float dA = __expf(dt * a[n]);
                h[n] = dA * h[n] + dtu * sBC[lt][n];
                y += h[n] * sBC[lt][D_STATE + n];
            }
            ys[idx] = y;
        }
        __syncthreads();
    }
}

// ---------------------------------------------------------------------------
// y = (ys + u * D) * silu(res), res = xr[..., 2048 + d]; in-place over ys
// ---------------------------------------------------------------------------
__global__ void combine_kernel(float* __restrict__ ys,
                               const float* __restrict__ u,
                               const float* __restrict__ Dvec,
                               const float* __restrict__ xr,
                               int total)
{
    const int i = blockIdx.x * blockDim.x + threadIdx.x;
    if (i >= total) return;
    const int d   = i & (D_INNER - 1);
    const int row = i >> 11;                 // / D_INNER
    const float res = xr[(size_t)row * (2 * D_INNER) + D_INNER + d];
    ys[i] = (ys[i] + u[i] * Dvec[d]) * silu_f(res);
}

// ---------------------------------------------------------------------------
// Launcher
// ---------------------------------------------------------------------------
extern "C" void kernel_launch(void* const* d_in, const int* in_sizes, int n_in,
                              void* d_out, int out_size, void* d_ws, size_t ws_size,
                              hipStream_t stream)
{
    (void)in_sizes; (void)n_in; (void)out_size; (void)ws_size;

    const float* x      = (const float*)d_in[0];
    const float* W_in   = (const float*)d_in[1];
    const float* conv_w = (const float*)d_in[2];
    const float* conv_b = (const float*)d_in[3];
    const float* W_x    = (const float*)d_in[4];
    const float* W_dt   = (const float*)d_in[5];
    const float* b_dt   = (const float*)d_in[6];
    const float* A_log  = (const float*)d_in[7];
    const float* Dvec   = (const float*)d_in[8];
    const float* W_out  = (const float*)d_in[9];

    // Workspace partition (floats):
    float* ws    = (float*)d_ws;
    float* xr    = ws;                                   // NROWS * 4096
    float* u     = xr    + (size_t)NROWS * (2 * D_INNER);// NROWS * 2048
    float* dbc   = u     + (size_t)NROWS * D_INNER;      // NROWS * 96
    float* delta = dbc   + (size_t)NROWS * 96;           // NROWS * 2048
    float* ys    = delta + (size_t)NROWS * D_INNER;      // NROWS * 2048

    const dim3 gblk(128);

    // 1) xr = x @ W_in : (4096 x 1024) @ (1024 x 4096)
    wmma_gemm_lds<4><<<dim3((2 * D_INNER) / 64, NROWS / 64), gblk, 0, stream>>>(
        x, W_in, xr, NROWS, 2 * D_INNER, D_MODEL, D_MODEL, 2 * D_INNER, 2 * D_INNER);

    // 2) depthwise conv + SiLU -> u
    conv_silu_kernel<<<dim3(D_INNER / 256, SEQLEN, BATCH), 256, 0, stream>>>(
        xr, conv_w, conv_b, u);

    // 3) dbc = u @ W_x : (4096 x 2048) @ (2048 x 96), N = 96 -> NT=2 (3 col blocks)
    wmma_gemm_lds<2><<<dim3((DT_RANK + 2 * D_STATE) / 32, NROWS / 64), gblk, 0, stream>>>(
        u, W_x, dbc, NROWS, DT_RANK + 2 * D_STATE, D_INNER,
        D_INNER, DT_RANK + 2 * D_STATE, DT_RANK + 2 * D_STATE);

    // 4) delta_pre = dbc[:, :64] @ W_dt : (4096 x 64) @ (64 x 2048)
    wmma_gemm_lds<4><<<dim3(D_INNER / 64, NROWS / 64), gblk, 0, stream>>>(
        dbc, W_dt, delta, NROWS, D_INNER, DT_RANK,
        DT_RANK + 2 * D_STATE, D_INNER, D_INNER);

    // 5) delta = softplus(delta_pre + b_dt)
    const int total = NROWS * D_INNER;
    softplus_bias_kernel<<<(total + 255) / 256, 256, 0, stream>>>(delta, b_dt, total);

    // 6) selective scan -> ys
    scan_kernel<<<dim3(D_INNER / 256, BATCH), 256, 0, stream>>>(
        delta, dbc, u, A_log, ys);

    // 7) y = (ys + u*D) * silu(res)   (in place over ys)
    combine_kernel<<<(total + 255) / 256, 256, 0, stream>>>(ys, u, Dvec, xr, total);

    // 8) out = y @ W_out : (4096 x 2048) @ (2048 x 1024)
    wmma_gemm_lds<4><<<dim3(D_MODEL / 64, NROWS / 64), gblk, 0, stream>>>(
        ys, W_out, (float*)d_out, NROWS, D_MODEL, D_INNER,
        D_INNER, D_MODEL, D_MODEL);
}